// NTEE_24558622998925
// MI455X (gfx1250) — compile-verified
//
#include <hip/hip_runtime.h>
#include <hip/hip_bf16.h>
#include <math.h>

#define BATCH    1024
#define TEXT_LEN 512
#define NUM_CAND 31
#define DIM      256

typedef __attribute__((ext_vector_type(2))) float v2f;
typedef __attribute__((ext_vector_type(8))) float v8f;

// ---------------------------------------------------------------------------
// Kernel 1: masked gather-sum over 512 tokens + L2 normalize.
// One block (64 threads = 2 waves) per sample; thread t owns dims [4t,4t+3]
// via float4, so each embedding row is read with 64 global_load_b128 lanes
// (one coalesced 1KB transaction, 4x fewer VMEM instructions than b32).
// ---------------------------------------------------------------------------
__global__ void __launch_bounds__(64) ntee_sum_norm(
    const float* __restrict__ word_emb,   // [WORD_VOCAB, 256]
    const int*   __restrict__ text_input, // [BATCH, 512]
    float*       __restrict__ vecn)       // [BATCH, 256] out
{
    __shared__ int   sIdx[TEXT_LEN];
    __shared__ float wsum[2];

    const int b    = blockIdx.x;
    const int t    = threadIdx.x;          // 0..63
    const int wave = t >> 5;
    const int lane = t & 31;

    // Stage the 512 token indices for this sample in LDS (uniform across block).
    #pragma unroll
    for (int i = 0; i < 8; ++i)
        sIdx[t + 64 * i] = text_input[b * TEXT_LEN + t + 64 * i];
    __syncthreads();

    const size_t dOff = (size_t)t * 4;     // first of 4 dims owned by this thread
    float4 acc = make_float4(0.f, 0.f, 0.f, 0.f);

    #pragma unroll 2
    for (int l = 0; l < TEXT_LEN; l += 4) {
        const int i0 = sIdx[l + 0];
        const int i1 = sIdx[l + 1];
        const int i2 = sIdx[l + 2];
        const int i3 = sIdx[l + 3];
        // Prefetch a future row to keep the memory pipe full.
        if (l + 16 < TEXT_LEN) {
            const int ip = sIdx[l + 16];
            if (ip != 0) __builtin_prefetch(word_emb + (size_t)ip * DIM + dOff, 0, 0);
        }
        // idx==0 is masked out in the reference; skipping the load is equivalent.
        if (i0 != 0) {
            const float4 v = *(const float4*)(word_emb + (size_t)i0 * DIM + dOff);
            acc.x += v.x; acc.y += v.y; acc.z += v.z; acc.w += v.w;
        }
        if (i1 != 0) {
            const float4 v = *(const float4*)(word_emb + (size_t)i1 * DIM + dOff);
            acc.x += v.x; acc.y += v.y; acc.z += v.z; acc.w += v.w;
        }
        if (i2 != 0) {
            const float4 v = *(const float4*)(word_emb + (size_t)i2 * DIM + dOff);
            acc.x += v.x; acc.y += v.y; acc.z += v.z; acc.w += v.w;
        }
        if (i3 != 0) {
            const float4 v = *(const float4*)(word_emb + (size_t)i3 * DIM + dOff);
            acc.x += v.x; acc.y += v.y; acc.z += v.z; acc.w += v.w;
        }
    }

    // Sum of squares: wave32 shuffle reduction, then combine the 2 waves in LDS.
    float sq = acc.x * acc.x + acc.y * acc.y + acc.z * acc.z + acc.w * acc.w;
    #pragma unroll
    for (int off = 16; off >= 1; off >>= 1)
        sq += __shfl_xor(sq, off, 32);
    if (lane == 0) wsum[wave] = sq;
    __syncthreads();

    const float inv_norm = 1.0f / sqrtf(wsum[0] + wsum[1]);
    float4 outv;
    outv.x = acc.x * inv_norm;
    outv.y = acc.y * inv_norm;
    outv.z = acc.z * inv_norm;
    outv.w = acc.w * inv_norm;
    *(float4*)(vecn + (size_t)b * DIM + dOff) = outv;
}

// ---------------------------------------------------------------------------
// Kernel 2: t = vecn @ W + b via v_wmma_f32_16x16x4_f32 (full f32 precision).
// One wave per 16x16 output tile; 8 waves/block, 128 blocks -> 64x16 tiles.
// K-loop: 64 WMMA steps of depth 4. EXEC is all-1s (no divergence).
// ---------------------------------------------------------------------------
__global__ void __launch_bounds__(256) ntee_affine_wmma(
    const float* __restrict__ vecn,  // [BATCH, 256]  (A)
    const float* __restrict__ W,     // [256, 256]    (B, row-major: W[k*256+n])
    const float* __restrict__ bias,  // [256]
    float*       __restrict__ tbuf)  // [BATCH, 256] out
{
    const int wave    = threadIdx.x >> 5;              // 0..7
    const int lane    = threadIdx.x & 31;              // wave32
    const int tile    = blockIdx.x * 8 + wave;         // 0..1023
    const int tileRow = tile >> 4;                     // 0..63  (batch/16)
    const int tileCol = tile & 15;                     // 0..15  (dim/16)
    const int m0      = tileRow * 16;
    const int n0      = tileCol * 16;

    const int r    = lane & 15;                        // row (A) / col (B) in tile
    const int half = lane >> 4;                        // K sub-slot selector

    // Bias-initialized accumulator: D row = v + 8*half, col = r.
    const float bval = bias[n0 + r];
    v8f acc;
    #pragma unroll
    for (int i = 0; i < 8; ++i) acc[i] = bval;

    const float* aRow = vecn + (size_t)(m0 + r) * DIM; // A row for this lane

    #pragma unroll 4
    for (int kb = 0; kb < DIM; kb += 4) {
        // A 16x4: lane holds A[r][kb+2*half], A[r][kb+2*half+1] -> one float2 load.
        v2f a = *(const v2f*)(aRow + kb + 2 * half);
        // B 4x16: lane holds W[kb+2*half][n0+r], W[kb+2*half+1][n0+r].
        v2f bb;
        bb.x = W[(size_t)(kb + 2 * half + 0) * DIM + n0 + r];
        bb.y = W[(size_t)(kb + 2 * half + 1) * DIM + n0 + r];
        acc = __builtin_amdgcn_wmma_f32_16x16x4_f32(
            /*neg_a=*/false, a, /*neg_b=*/false, bb,
            /*c_mod=*/(short)0, acc, /*reuse_a=*/false, /*reuse_b=*/false);
    }

    // Store D: VGPR v -> row m0 + v + 8*half, col n0 + r.
    #pragma unroll
    for (int v = 0; v < 8; ++v)
        tbuf[(size_t)(m0 + v + 8 * half) * DIM + n0 + r] = acc[v];
}

// ---------------------------------------------------------------------------
// Kernel 3: sim[b,k] = t[b] . entity_emb[entity_input[b,k]], softmax over 31.
// One block per sample; each wave handles 4 candidates. Each lane covers 8
// dims via two b128 loads, then a wave32 shuffle reduction.
// ---------------------------------------------------------------------------
__global__ void __launch_bounds__(256) ntee_sim_softmax(
    const float* __restrict__ entity_emb,   // [ENT_VOCAB, 256]
    const int*   __restrict__ entity_input, // [BATCH, 31]
    const float* __restrict__ tbuf,         // [BATCH, 256]
    float*       __restrict__ out)          // [BATCH, 31]
{
    __shared__ float tvec[DIM];
    __shared__ float sims[32];

    const int b    = blockIdx.x;
    const int tid  = threadIdx.x;
    const int wave = tid >> 5;
    const int lane = tid & 31;

    tvec[tid] = tbuf[(size_t)b * DIM + tid];
    if (tid < 32) sims[tid] = -INFINITY;
    __syncthreads();

    const int d0 = lane * 8;
    // t-vector slice for this lane: two ds_load_b128.
    const float4 t0 = *(const float4*)(tvec + d0);
    const float4 t1 = *(const float4*)(tvec + d0 + 4);

    #pragma unroll
    for (int i = 0; i < 4; ++i) {
        const int k = wave + 8 * i;
        if (k < NUM_CAND) {
            const int eidx = entity_input[b * NUM_CAND + k];
            const float* erow = entity_emb + (size_t)eidx * DIM + d0;
            const float4 e0 = *(const float4*)(erow);
            const float4 e1 = *(const float4*)(erow + 4);
            float partial = e0.x * t0.x + e0.y * t0.y + e0.z * t0.z + e0.w * t0.w
                          + e1.x * t1.x + e1.y * t1.y + e1.z * t1.z + e1.w * t1.w;
            // wave32 reduction
            #pragma unroll
            for (int off = 16; off >= 1; off >>= 1)
                partial += __shfl_xor(partial, off, 32);
            if (lane == 0) sims[k] = partial;
        }
    }
    __syncthreads();

    // Softmax over 31 candidates, done by wave 0 (slot 31 padded with -inf).
    if (tid < 32) {
        const float s = sims[tid];
        float m = s;
        #pragma unroll
        for (int off = 16; off >= 1; off >>= 1)
            m = fmaxf(m, __shfl_xor(m, off, 32));
        const float e = (tid < NUM_CAND) ? __expf(s - m) : 0.0f;
        float sum = e;
        #pragma unroll
        for (int off = 16; off >= 1; off >>= 1)
            sum += __shfl_xor(sum, off, 32);
        if (tid < NUM_CAND) out[b * NUM_CAND + tid] = e / sum;
    }
}

// ---------------------------------------------------------------------------
extern "C" void kernel_launch(void* const* d_in, const int* in_sizes, int n_in,
                              void* d_out, int out_size, void* d_ws, size_t ws_size,
                              hipStream_t stream) {
    const float* word_emb     = (const float*)d_in[0]; // [200000, 256]
    const float* entity_emb   = (const float*)d_in[1]; // [300000, 256]
    const float* W            = (const float*)d_in[2]; // [256, 256]
    const float* bvec         = (const float*)d_in[3]; // [256]
    const int*   text_input   = (const int*)d_in[4];   // [1024, 512]
    const int*   entity_input = (const int*)d_in[5];   // [1024, 31]
    // d_in[6] labels: unused (reference ignores them).
    float* out = (float*)d_out;                        // [1024, 31]

    float* vecn = (float*)d_ws;                        // [1024, 256]
    float* tbuf = vecn + (size_t)BATCH * DIM;          // [1024, 256]

    ntee_sum_norm<<<BATCH, 64, 0, stream>>>(word_emb, text_input, vecn);
    ntee_affine_wmma<<<BATCH / 8, 256, 0, stream>>>(vecn, W, bvec, tbuf);
    ntee_sim_softmax<<<BATCH, 256, 0, stream>>>(entity_emb, entity_input, tbuf, out);
}